// LightweightConv1dTBC_11081015624096
// MI455X (gfx1250) — compile-verified
//
#include <hip/hip_runtime.h>
#include <hip/hip_bf16.h>

// Problem constants (from reference): T=2048, B=2, C=1024, H=16, K=31, P=15
#define TT_   2048
#define BB_   2
#define CC_   1024
#define BC_   (BB_ * CC_)   // 2048
#define HH_   16
#define RR_   (CC_ / HH_)   // 64
#define KK_   31
#define PP_   15

typedef float v2f __attribute__((ext_vector_type(2)));
typedef float v4f __attribute__((ext_vector_type(4)));
typedef float v8f __attribute__((ext_vector_type(8)));

// ---------------------------------------------------------------------------
// Kernel 1: per-head softmax of raw weights (H=16 rows of K=31).
// One wave (32 lanes) per head; result stored padded to stride 32 in d_ws
// (slot 31 of each row holds 0.0 so masked/wrapped lookups read zero).
// ---------------------------------------------------------------------------
__global__ void lwconv_softmax_kernel(const float* __restrict__ wraw,
                                      float* __restrict__ wsw) {
    const int h    = threadIdx.x >> 5;   // 16 waves in a 512-thread block
    const int lane = threadIdx.x & 31;

    float v = (lane < KK_) ? wraw[h * KK_ + lane] : -3.402823466e38f;

    float m = v;
    #pragma unroll
    for (int o = 16; o > 0; o >>= 1) m = fmaxf(m, __shfl_xor(m, o, 32));

    float e = (lane < KK_) ? expf(v - m) : 0.0f;

    float s = e;
    #pragma unroll
    for (int o = 16; o > 0; o >>= 1) s += __shfl_xor(s, o, 32);

    wsw[h * 32 + lane] = e / s;          // lane 31 writes 0
}

// ---------------------------------------------------------------------------
// Kernel 2: depthwise conv as banded-Toeplitz matmul via V_WMMA_F32_16X16X4_F32.
//   Out(T x BC) = Band_h(T x T, bandwidth 31) x X(T x BC)
// One wave computes one 16x16 output tile; 12 K-steps of 4 cover
// t' in [t0-16, t0+32) which contains the full band [t0-15, t0+30].
//
// All loads are branchless: indices are wrapped into valid ranges (&-mask)
// and values zeroed with a cndmask-selected 0/1 multiplier, so the wave
// never manipulates EXEC and the load stream pipelines ahead of the WMMAs.
//
// Fragment layouts (CDNA5 ISA 7.12.2, wave32):
//   A 16x4 f32 : M = lane%16 ; VGPR0/VGPR1 = K {0,1} (lanes 0-15) | {2,3} (16-31)
//   B 4x16 f32 : N = lane%16 ; VGPR0/VGPR1 = K {0,1} (lanes 0-15) | {2,3} (16-31)
//   C/D 16x16  : N = lane%16 ; VGPR v = M=v (lanes 0-15) | M=v+8 (lanes 16-31)
// ---------------------------------------------------------------------------
__global__ __launch_bounds__(256) void lwconv_wmma_kernel(
        const float* __restrict__ x,     // (T, B, C) flat
        const float* __restrict__ wsw,   // (H, 32) softmaxed weights
        float* __restrict__ out1) {      // (T, B, C) flat

    const int lane = threadIdx.x & 31;
    const int wid  = blockIdx.x * (blockDim.x >> 5) + (threadIdx.x >> 5);

    const int tt = wid >> 7;             // t-tile   [0, 128)
    const int ct = wid & 127;            // col-tile [0, 128)
    const int t0 = tt << 4;

    const int n     = lane & 15;
    const int col   = (ct << 4) + n;     // flat column in the (B*C) plane
    const int c     = col & (CC_ - 1);
    const int h     = c >> 6;            // c / R, R = 64
    const int khalf = (lane >> 4) << 1;  // 0 for lanes 0-15, 2 for lanes 16-31
    const int i     = lane & 15;         // A-matrix row (M)

    const float* __restrict__ wrow = wsw + h * 32;

    // Pre-gather all 24 A-weights and 24 B-x values branchlessly.
    float aw[24];
    float bx[24];
    const int baset = t0 - 16;

    #pragma unroll
    for (int s = 0; s < 12; ++s) {
        const int kk0 = 4 * s + khalf;

        // A: Band[t0+i][baset+kk] = w[h, kk - i - 1]; wrap into padded row,
        // slot 31 is 0.0 and the mask kills truly-out-of-range offsets.
        const int off0 = kk0 - i - 1;
        const int off1 = off0 + 1;
        const float ma0 = ((unsigned)off0 < (unsigned)KK_) ? 1.0f : 0.0f;
        const float ma1 = ((unsigned)off1 < (unsigned)KK_) ? 1.0f : 0.0f;
        aw[2 * s + 0] = wrow[off0 & 31] * ma0;
        aw[2 * s + 1] = wrow[off1 & 31] * ma1;

        // B: X[baset+kk][col]; wrap t' into [0,T) (always a valid address),
        // mask zeroes the halo rows.
        const int tp0 = baset + kk0;
        const int tp1 = tp0 + 1;
        const float mb0 = ((unsigned)tp0 < (unsigned)TT_) ? 1.0f : 0.0f;
        const float mb1 = ((unsigned)tp1 < (unsigned)TT_) ? 1.0f : 0.0f;
        bx[2 * s + 0] = x[(size_t)(tp0 & (TT_ - 1)) * BC_ + col] * mb0;
        bx[2 * s + 1] = x[(size_t)(tp1 & (TT_ - 1)) * BC_ + col] * mb1;
    }

    v8f acc = {};
    #pragma unroll
    for (int s = 0; s < 12; ++s) {
        v2f a; a.x = aw[2 * s + 0]; a.y = aw[2 * s + 1];
        v2f b; b.x = bx[2 * s + 0]; b.y = bx[2 * s + 1];
        acc = __builtin_amdgcn_wmma_f32_16x16x4_f32(
            /*neg_a=*/false, a, /*neg_b=*/false, b,
            /*c_mod=*/(short)0, acc, /*reuse_a=*/false, /*reuse_b=*/false);
    }

    const int mbase = (lane < 16) ? 0 : 8;
    #pragma unroll
    for (int v = 0; v < 8; ++v) {
        out1[(size_t)(t0 + mbase + v) * BC_ + col] = acc[v];
    }
}

// ---------------------------------------------------------------------------
// Kernel 3: weight_expanded (B*H, T, T) banded Toeplitz fill.
// 537 MB of stores -> float4 non-temporal streaming. The band test is a
// single wave-coherent branch (a wave spans 128 consecutive j in one row),
// so ~92% of waves take the pure zero-store path with no table loads.
// ---------------------------------------------------------------------------
__global__ __launch_bounds__(256) void lwconv_band_kernel(
        const float* __restrict__ wsw,
        float* __restrict__ out2) {

    const long idx = (long)blockIdx.x * blockDim.x + threadIdx.x; // float4 index
    const int  j4  = (int)(idx & (TT_ / 4 - 1));                  // 512 float4 / row
    const long r   = idx >> 9;
    const int  i   = (int)(r & (TT_ - 1));
    const int  bh  = (int)(r >> 11);
    const int  h   = bh & (HH_ - 1);

    const int offlo = (j4 << 2) - i + PP_;   // off for element u=0 of this float4

    v4f v = {0.0f, 0.0f, 0.0f, 0.0f};
    if (offlo >= -3 && offlo <= KK_ - 1) {   // float4 intersects the band
        const float* __restrict__ wrow = wsw + h * 32;
        #pragma unroll
        for (int u = 0; u < 4; ++u) {
            const int off = offlo + u;
            v[u] = ((unsigned)off < (unsigned)KK_) ? wrow[off] : 0.0f;
        }
    }

    __builtin_nontemporal_store(v, (v4f*)out2 + idx);
}

// ---------------------------------------------------------------------------
// Launch
// ---------------------------------------------------------------------------
extern "C" void kernel_launch(void* const* d_in, const int* in_sizes, int n_in,
                              void* d_out, int out_size, void* d_ws, size_t ws_size,
                              hipStream_t stream) {
    const float* x    = (const float*)d_in[0];   // (T, B, C)
    const float* wraw = (const float*)d_in[1];   // (H, 1, K)

    float* out  = (float*)d_out;
    float* out1 = out;                                    // (T, B, C)
    float* out2 = out + (size_t)TT_ * BB_ * CC_;          // (B*H, T, T)
    float* wsw  = (float*)d_ws;                           // H*32 floats

    // 1) softmax weights: 1 block, 16 waves (one per head)
    lwconv_softmax_kernel<<<1, HH_ * 32, 0, stream>>>(wraw, wsw);

    // 2) conv via WMMA: 128*128 tiles, 8 waves/block -> 2048 blocks
    lwconv_wmma_kernel<<<(TT_ / 16) * (BC_ / 16) / 8, 256, 0, stream>>>(x, wsw, out1);

    // 3) Toeplitz band fill: 32*2048*512 float4 NT stores
    const long n4 = (long)BB_ * HH_ * TT_ * (TT_ / 4);    // 33,554,432
    lwconv_band_kernel<<<(unsigned)(n4 / 256), 256, 0, stream>>>(wsw, out2);
}